// RandomWalkKpmNet_11312943857861
// MI455X (gfx1250) — compile-verified
//
#include <hip/hip_runtime.h>
#include <math.h>

#define G_DIM 3072
#define P_DIM 1024
#define ALPHA 0.1f
#define NITER 8   // S = sum_{k=0}^{9} (alpha*A)^k ; tail < 1.2e-10

typedef float v2f __attribute__((ext_vector_type(2)));
typedef float v8f __attribute__((ext_vector_type(8)));

// ---------------------------------------------------------------------------
// B = alpha * softmax(row of g_g_score[:,:,1] with diagonal = -inf)
// one 256-thread block per row; G/256 = 12 elements per thread.
// ---------------------------------------------------------------------------
__global__ __launch_bounds__(256) void build_B_kernel(const float* __restrict__ gg,
                                                      float* __restrict__ B) {
    const int row = blockIdx.x;
    const int t = threadIdx.x;
    __shared__ float red[256];

    float xv[12];
#pragma unroll
    for (int i = 0; i < 12; ++i) {
        int j = t + i * 256;
        xv[i] = (j == row) ? -INFINITY
                           : gg[((size_t)row * G_DIM + j) * 2 + 1];
    }
    // max reduction
    float m = -INFINITY;
#pragma unroll
    for (int i = 0; i < 12; ++i) m = fmaxf(m, xv[i]);
    red[t] = m;
    __syncthreads();
    for (int off = 128; off > 0; off >>= 1) {
        if (t < off) red[t] = fmaxf(red[t], red[t + off]);
        __syncthreads();
    }
    m = red[0];
    __syncthreads();
    // sum of exp
    float s = 0.f;
#pragma unroll
    for (int i = 0; i < 12; ++i) s += __expf(xv[i] - m);
    red[t] = s;
    __syncthreads();
    for (int off = 128; off > 0; off >>= 1) {
        if (t < off) red[t] += red[t + off];
        __syncthreads();
    }
    s = red[0];
    const float scale = ALPHA / s;
#pragma unroll
    for (int i = 0; i < 12; ++i) {
        int j = t + i * 256;
        B[(size_t)row * G_DIM + j] = __expf(xv[i] - m) * scale;
    }
}

// ---------------------------------------------------------------------------
// S0 = B + I
// ---------------------------------------------------------------------------
__global__ __launch_bounds__(256) void init_S_kernel(const float* __restrict__ B,
                                                     float* __restrict__ S) {
    size_t idx = (size_t)blockIdx.x * 256 + threadIdx.x;
    int r = (int)(idx / G_DIM);
    int c = (int)(idx - (size_t)r * G_DIM);
    float v = B[idx];
    if (r == c) v += 1.0f;
    S[idx] = v;
}

// ---------------------------------------------------------------------------
// Horner GEMM: Sout = Bm * Sin + I      (all G x G, row-major, f32)
// block tile 128x128, 8 waves (wave32) in 2x4, wave tile 64x32.
// ---------------------------------------------------------------------------
__global__ __launch_bounds__(256) void gemm_horner(const float* __restrict__ Bm,
                                                   const float* __restrict__ Sin,
                                                   float* __restrict__ Sout) {
    __shared__ float a_lds[128 * 16];   // [m][k]
    __shared__ float b_lds[128 * 16];   // [n][k] (transposed stage)

    const int NB = G_DIM / 128;
    const int bm = blockIdx.x / NB;
    const int bn = blockIdx.x % NB;
    const int m_base = bm * 128;
    const int n_base = bn * 128;

    const int t = threadIdx.x;
    const int wave = t >> 5;
    const int lane = t & 31;
    const int wm0 = (wave & 1) * 64;    // wave M offset inside block tile
    const int wn0 = (wave >> 1) * 32;   // wave N offset inside block tile

    v8f acc[4][2];
#pragma unroll
    for (int mt = 0; mt < 4; ++mt)
#pragma unroll
        for (int nt = 0; nt < 2; ++nt) acc[mt][nt] = (v8f)(0.0f);

    const int lr = t >> 1;            // 0..127 : A row in tile
    const int lk = (t & 1) * 8;       // 0 or 8 : A k half
    const int kb = t >> 4;            // 0..15  : B k line
    const int nb8 = (t & 15) * 8;     // 0..120 : B n start

    for (int k0 = 0; k0 < G_DIM; k0 += 16) {
        // stage A tile [128m x 16k]
        const float* ga = Bm + (size_t)(m_base + lr) * G_DIM + k0 + lk;
        *(float4*)&a_lds[lr * 16 + lk]     = *(const float4*)ga;
        *(float4*)&a_lds[lr * 16 + lk + 4] = *(const float4*)(ga + 4);
        // stage B tile: B[k][n] = Sin[(k0+k)*G + n_base+n], store as [n][k]
        const float* gb = Sin + (size_t)(k0 + kb) * G_DIM + n_base + nb8;
#pragma unroll
        for (int i = 0; i < 8; ++i) b_lds[(nb8 + i) * 16 + kb] = gb[i];
        __syncthreads();

#pragma unroll
        for (int kk = 0; kk < 16; kk += 4) {
            const int krd = kk + ((lane >> 4) << 1);
            v2f afr[4], bfr[2];
#pragma unroll
            for (int mt = 0; mt < 4; ++mt)
                afr[mt] = *(const v2f*)&a_lds[(wm0 + mt * 16 + (lane & 15)) * 16 + krd];
#pragma unroll
            for (int nt = 0; nt < 2; ++nt)
                bfr[nt] = *(const v2f*)&b_lds[(wn0 + nt * 16 + (lane & 15)) * 16 + krd];
#pragma unroll
            for (int mt = 0; mt < 4; ++mt)
#pragma unroll
                for (int nt = 0; nt < 2; ++nt)
                    acc[mt][nt] = __builtin_amdgcn_wmma_f32_16x16x4_f32(
                        false, afr[mt], false, bfr[nt], (short)0, acc[mt][nt],
                        false, false);
        }
        __syncthreads();
    }

    // epilogue: + identity
#pragma unroll
    for (int mt = 0; mt < 4; ++mt) {
#pragma unroll
        for (int nt = 0; nt < 2; ++nt) {
            const int col = n_base + wn0 + nt * 16 + (lane & 15);
#pragma unroll
            for (int v = 0; v < 8; ++v) {
                const int row = m_base + wm0 + mt * 16 + v + ((lane >> 4) << 3);
                float val = acc[mt][nt][v];
                if (row == col) val += 1.0f;
                Sout[(size_t)row * G_DIM + col] = val;
            }
        }
    }
}

// ---------------------------------------------------------------------------
// Final apply: Out[r][h] = 0.9 * sum_k X[r][k] * S[h][k]   (i.e. X @ S^T)
// X rows 0..1023: pg ch0; 1024..2047: pg ch1; 2048..5119: gg ch0; 5120..: gg ch1
// Output scattered back into the interleaved [.,.,2] layout.
// ---------------------------------------------------------------------------
__device__ __forceinline__ float loadX(const float* __restrict__ pg,
                                       const float* __restrict__ gg,
                                       int r, int k) {
    if (r < P_DIM)            return pg[((size_t)r * G_DIM + k) * 2];
    if (r < 2 * P_DIM)        return pg[((size_t)(r - P_DIM) * G_DIM + k) * 2 + 1];
    if (r < 2 * P_DIM + G_DIM)return gg[((size_t)(r - 2 * P_DIM) * G_DIM + k) * 2];
    return gg[((size_t)(r - 2 * P_DIM - G_DIM) * G_DIM + k) * 2 + 1];
}

__global__ __launch_bounds__(256) void gemm_final(const float* __restrict__ pg,
                                                  const float* __restrict__ gg,
                                                  const float* __restrict__ S,
                                                  float* __restrict__ out) {
    __shared__ float a_lds[128 * 16];   // [r][k]
    __shared__ float b_lds[128 * 16];   // [h][k]  (S^T tile == S rows)

    const int NB = G_DIM / 128;
    const int bm = blockIdx.x / NB;
    const int bn = blockIdx.x % NB;
    const int m_base = bm * 128;
    const int n_base = bn * 128;

    const int t = threadIdx.x;
    const int wave = t >> 5;
    const int lane = t & 31;
    const int wm0 = (wave & 1) * 64;
    const int wn0 = (wave >> 1) * 32;

    v8f acc[4][2];
#pragma unroll
    for (int mt = 0; mt < 4; ++mt)
#pragma unroll
        for (int nt = 0; nt < 2; ++nt) acc[mt][nt] = (v8f)(0.0f);

    const int lr = t >> 1;
    const int lk = (t & 1) * 8;

    for (int k0 = 0; k0 < G_DIM; k0 += 16) {
        // A tile: strided 2-channel gather from the original tensors
#pragma unroll
        for (int i = 0; i < 8; ++i)
            a_lds[lr * 16 + lk + i] = loadX(pg, gg, m_base + lr, k0 + lk + i);
        // B tile: B[k][h] = S[n_base+h][k0+k] -> contiguous along k
        const float* gb = S + (size_t)(n_base + lr) * G_DIM + k0 + lk;
        *(float4*)&b_lds[lr * 16 + lk]     = *(const float4*)gb;
        *(float4*)&b_lds[lr * 16 + lk + 4] = *(const float4*)(gb + 4);
        __syncthreads();

#pragma unroll
        for (int kk = 0; kk < 16; kk += 4) {
            const int krd = kk + ((lane >> 4) << 1);
            v2f afr[4], bfr[2];
#pragma unroll
            for (int mt = 0; mt < 4; ++mt)
                afr[mt] = *(const v2f*)&a_lds[(wm0 + mt * 16 + (lane & 15)) * 16 + krd];
#pragma unroll
            for (int nt = 0; nt < 2; ++nt)
                bfr[nt] = *(const v2f*)&b_lds[(wn0 + nt * 16 + (lane & 15)) * 16 + krd];
#pragma unroll
            for (int mt = 0; mt < 4; ++mt)
#pragma unroll
                for (int nt = 0; nt < 2; ++nt)
                    acc[mt][nt] = __builtin_amdgcn_wmma_f32_16x16x4_f32(
                        false, afr[mt], false, bfr[nt], (short)0, acc[mt][nt],
                        false, false);
        }
        __syncthreads();
    }

    // epilogue: scale by (1-alpha) and scatter to interleaved output
#pragma unroll
    for (int mt = 0; mt < 4; ++mt) {
#pragma unroll
        for (int nt = 0; nt < 2; ++nt) {
            const int col = n_base + wn0 + nt * 16 + (lane & 15);
#pragma unroll
            for (int v = 0; v < 8; ++v) {
                const int row = m_base + wm0 + mt * 16 + v + ((lane >> 4) << 3);
                const float val = 0.9f * acc[mt][nt][v];
                size_t o;
                if (row < P_DIM)
                    o = ((size_t)row * G_DIM + col) * 2;
                else if (row < 2 * P_DIM)
                    o = ((size_t)(row - P_DIM) * G_DIM + col) * 2 + 1;
                else if (row < 2 * P_DIM + G_DIM)
                    o = (size_t)P_DIM * G_DIM * 2 +
                        ((size_t)(row - 2 * P_DIM) * G_DIM + col) * 2;
                else
                    o = (size_t)P_DIM * G_DIM * 2 +
                        ((size_t)(row - 2 * P_DIM - G_DIM) * G_DIM + col) * 2 + 1;
                out[o] = val;
            }
        }
    }
}

// ---------------------------------------------------------------------------
extern "C" void kernel_launch(void* const* d_in, const int* in_sizes, int n_in,
                              void* d_out, int out_size, void* d_ws, size_t ws_size,
                              hipStream_t stream) {
    const float* pg = (const float*)d_in[0];   // [P, G, 2]
    const float* gg = (const float*)d_in[1];   // [G, G, 2]
    float* out = (float*)d_out;

    const size_t mat = (size_t)G_DIM * G_DIM;
    float* Bm = (float*)d_ws;       // alpha * softmax adjacency
    float* Sa = Bm + mat;           // Neumann-series ping
    float* Sb = Sa + mat;           // Neumann-series pong

    build_B_kernel<<<G_DIM, 256, 0, stream>>>(gg, Bm);
    init_S_kernel<<<(int)(mat / 256), 256, 0, stream>>>(Bm, Sa);

    const int gemm_blocks = (G_DIM / 128) * (G_DIM / 128);   // 576
    float* cur = Sa;
    float* nxt = Sb;
    for (int it = 0; it < NITER; ++it) {
        gemm_horner<<<gemm_blocks, 256, 0, stream>>>(Bm, cur, nxt);
        float* tmp = cur; cur = nxt; nxt = tmp;
    }
    // after an even number of swaps the final series sits in Sa (== cur)
    const int final_blocks = ((2 * (P_DIM + G_DIM)) / 128) * (G_DIM / 128); // 1536
    gemm_final<<<final_blocks, 256, 0, stream>>>(pg, gg, cur, out);
}